// PhysQuadModel_26963804685107
// MI455X (gfx1250) — compile-verified
//
#include <hip/hip_runtime.h>
#include <stdint.h>
#include <math.h>

#define BLOCK 128
#define TS 4

#define DT_    0.01f
#define MASS_  0.033f
#define G_     9.81f
#define KT_    3.72e-08f
#define TMAX_  (1.8f * 0.033f * 9.81f)
#define EPS_   1e-06f

// ---------- CDNA5 async global <-> LDS DMA (ASYNCcnt-tracked) ----------
__device__ __forceinline__ unsigned lds_off32(const void* p) {
    // generic LDS address: addr[31:0] is the LDS byte offset (ISA 10.2)
    return (unsigned)(uintptr_t)p;
}
__device__ __forceinline__ void async_load_b128(unsigned lds_addr, const void* gaddr) {
    asm volatile("global_load_async_to_lds_b128 %0, %1, off"
                 :: "v"(lds_addr), "v"(gaddr) : "memory");
}
__device__ __forceinline__ void async_store_b128(const void* gaddr, unsigned lds_addr) {
    asm volatile("global_store_async_from_lds_b128 %0, %1, off"
                 :: "v"(gaddr), "v"(lds_addr) : "memory");
}
__device__ __forceinline__ void wait_async0() {
    asm volatile("s_wait_asynccnt 0" ::: "memory");
}

// ---------- quaternion helpers (match reference math; omega_dot == 0) ----------
struct Quat { float x, y, z, w; };

__device__ __forceinline__ Quat qnorm(Quat q) {
    float n = sqrtf(q.x*q.x + q.y*q.y + q.z*q.z + q.w*q.w);
    float inv = 1.0f / fmaxf(n, 1e-12f);
    return {q.x*inv, q.y*inv, q.z*inv, q.w*inv};
}
__device__ __forceinline__ Quat qaxpy(Quat q, float s, Quat d) {
    return {q.x + s*d.x, q.y + s*d.y, q.z + s*d.z, q.w + s*d.w};
}
// quat_derivative with constant omega
__device__ __forceinline__ Quat qdot_(Quat q, float ox, float oy, float oz) {
    Quat d;
    d.x = 0.5f * (q.w*ox + (q.y*oz - q.z*oy));
    d.y = 0.5f * (q.w*oy + (q.z*ox - q.x*oz));
    d.z = 0.5f * (q.w*oz + (q.x*oy - q.y*ox));
    d.w = -0.5f * (q.x*ox + q.y*oy + q.z*oz);
    return d;
}
// acc = quat_rotate(q, (0,0,T))/M - (0,0,G)
__device__ __forceinline__ void thrust_acc(Quat q, float T,
                                           float& ax, float& ay, float& az) {
    float tx =  2.0f * q.y * T;
    float ty = -2.0f * q.x * T;
    float wx = q.w*tx - q.z*ty;
    float wy = q.w*ty + q.z*tx;
    float wz = T + (q.x*ty - q.y*tx);
    const float invM = 1.0f / MASS_;
    ax = wx * invM;
    ay = wy * invM;
    az = wz * invM - G_;
}

__global__ __launch_bounds__(BLOCK)
void quad_rollout_kernel(const float* __restrict__ x0,
                         const float* __restrict__ u_seq,
                         float* __restrict__ out, int N)
{
    // u stage: [2 buffers][TS][BLOCK rows][4 floats]  (double-buffered, 16 KB)
    __shared__ __align__(16) float su[2][TS * BLOCK * 4];
    // out stage: [BLOCK rows][52 dwords] (TS*12 data + 4 pad dwords) (26 KB)
    __shared__ __align__(16) float so_[BLOCK * 52];

    const int tid = threadIdx.x;
    const int b0  = blockIdx.x * BLOCK;
    const int b   = b0 + tid;

    // load initial state: 12 floats per row, 48B-aligned
    const float4* xp = (const float4*)(x0 + (size_t)b * 12);
    float4 A = xp[0], Bv = xp[1], C = xp[2];
    float px = A.x, py = A.y, pz = A.z;
    float vx = A.w, vy = Bv.x, vz = Bv.y;
    float rx = Bv.z, ry = Bv.w, rz = C.x;   // so3 state
    const float ox = C.y, oy = C.z, oz = C.w; // omega (constant: omega_dot==0)

    const unsigned su_base = lds_off32(&su[0][0]);
    const unsigned so_base = lds_off32(&so_[0]);
    const int nchunks = N / TS;

    // prologue: async-load chunk 0 into buffer 0 (coalesced 16B/lane chunks)
    #pragma unroll
    for (int j = 0; j < TS; ++j) {
        int k   = j * BLOCK + tid;
        int row = k >> 2;       // k / TS
        int t   = k & 3;        // k % TS  (one 16B chunk == u[b][t][0:4])
        const void* g = (const char*)u_seq + (((size_t)(b0 + row)) * N + t) * 16;
        async_load_b128(su_base + (unsigned)((t * BLOCK + row) * 16), g);
    }

    for (int c = 0; c < nchunks; ++c) {
        wait_async0();       // loads for chunk c ready; prior stores drained
        __syncthreads();

        const int cur = c & 1;
        if (c + 1 < nchunks) {
            // prefetch next chunk into the other buffer (overlaps compute)
            int t0n = (c + 1) * TS;
            unsigned nb = su_base + (unsigned)(((c + 1) & 1) * (TS * BLOCK * 16));
            #pragma unroll
            for (int j = 0; j < TS; ++j) {
                int k   = j * BLOCK + tid;
                int row = k >> 2;
                int t   = k & 3;
                const void* g = (const char*)u_seq +
                                (((size_t)(b0 + row)) * N + (t0n + t)) * 16;
                async_load_b128(nb + (unsigned)((t * BLOCK + row) * 16), g);
            }
        }

        #pragma unroll
        for (int t = 0; t < TS; ++t) {
            const float4 u = *(const float4*)&su[cur][(t * BLOCK + tid) * 4];

            // motor_to_phys -> thrust only (torques feed omega_dot which is *0)
            float ssum = u.x*u.x + u.y*u.y + u.z*u.z + u.w*u.w;
            float Tn = (KT_ * ssum) * (1.0f / TMAX_);
            Tn = fminf(fmaxf(Tn, 0.0f), 1.0f);
            const float T = Tn * TMAX_;

            // so3 -> quat (reference small-angle handling)
            float ang  = sqrtf(rx*rx + ry*ry + rz*rz);
            float half = 0.5f * ang;
            bool  sm   = ang < EPS_;
            float sov  = sm ? (0.5f - ang*ang*(1.0f/48.0f)) : (sinf(half)/ang);
            Quat q = { rx*sov, ry*sov, rz*sov, cosf(half) };

            // RK4 (omega constant -> only quat/vel/pos evolve)
            float a1x,a1y,a1z, a2x,a2y,a2z, a3x,a3y,a3z, a4x,a4y,a4z;
            thrust_acc(q, T, a1x, a1y, a1z);
            Quat qd1 = qdot_(q, ox, oy, oz);
            Quat q2  = qnorm(qaxpy(q, 0.5f*DT_, qd1));
            thrust_acc(q2, T, a2x, a2y, a2z);
            Quat qd2 = qdot_(q2, ox, oy, oz);
            Quat q3  = qnorm(qaxpy(q, 0.5f*DT_, qd2));
            thrust_acc(q3, T, a3x, a3y, a3z);
            Quat qd3 = qdot_(q3, ox, oy, oz);
            Quat q4  = qnorm(qaxpy(q, DT_, qd3));
            thrust_acc(q4, T, a4x, a4y, a4z);
            Quat qd4 = qdot_(q4, ox, oy, oz);

            const float c6 = DT_ / 6.0f;
            // pos: v1=vel, v2=vel+.5dt*a1, v3=vel+.5dt*a2, v4=vel+dt*a3
            px += c6 * (vx + 2.0f*(vx + 0.5f*DT_*a1x) + 2.0f*(vx + 0.5f*DT_*a2x) + (vx + DT_*a3x));
            py += c6 * (vy + 2.0f*(vy + 0.5f*DT_*a1y) + 2.0f*(vy + 0.5f*DT_*a2y) + (vy + DT_*a3y));
            pz += c6 * (vz + 2.0f*(vz + 0.5f*DT_*a1z) + 2.0f*(vz + 0.5f*DT_*a2z) + (vz + DT_*a3z));
            vx += c6 * (a1x + 2.0f*a2x + 2.0f*a3x + a4x);
            vy += c6 * (a1y + 2.0f*a2y + 2.0f*a3y + a4y);
            vz += c6 * (a1z + 2.0f*a2z + 2.0f*a3z + a4z);
            Quat qn;
            qn.x = q.x + c6*(qd1.x + 2.0f*qd2.x + 2.0f*qd3.x + qd4.x);
            qn.y = q.y + c6*(qd1.y + 2.0f*qd2.y + 2.0f*qd3.y + qd4.y);
            qn.z = q.z + c6*(qd1.z + 2.0f*qd2.z + 2.0f*qd3.z + qd4.z);
            qn.w = q.w + c6*(qd1.w + 2.0f*qd2.w + 2.0f*qd3.w + qd4.w);
            qn = qnorm(qn);

            // quat -> so3
            float nv = sqrtf(qn.x*qn.x + qn.y*qn.y + qn.z*qn.z);
            float hf = atan2f(nv, qn.w);
            float an = 2.0f * hf;
            bool  sm2 = fabsf(an) < EPS_;
            float so2 = sm2 ? (0.5f - an*an*(1.0f/48.0f)) : (sinf(hf)/an);
            float inv = 1.0f / so2;
            rx = qn.x * inv; ry = qn.y * inv; rz = qn.z * inv;

            // stage 12-float state row into LDS (16B aligned: 208B row stride)
            float* orow = &so_[tid * 52 + t * 12];
            ((float4*)orow)[0] = make_float4(px, py, pz, vx);
            ((float4*)orow)[1] = make_float4(vy, vz, rx, ry);
            ((float4*)orow)[2] = make_float4(rz, ox, oy, oz);
        }
        __syncthreads();   // all rows of so_ finalized (cross-wave)

        // async-store chunk c: coalesced 16B chunks, 12 per row (TS*48B)
        {
            const int t0 = c * TS;
            #pragma unroll
            for (int j = 0; j < 12; ++j) {
                int k   = j * BLOCK + tid;
                int row = k / 12;
                int m   = k - row * 12;
                const void* g = (const char*)out +
                                (((size_t)(b0 + row)) * N + t0) * 48 + (size_t)m * 16;
                async_store_b128(g, so_base + (unsigned)(row * 208 + m * 16));
            }
        }
    }
    wait_async0();  // drain final stores (S_ENDPGM also waits idle)
}

extern "C" void kernel_launch(void* const* d_in, const int* in_sizes, int n_in,
                              void* d_out, int out_size, void* d_ws, size_t ws_size,
                              hipStream_t stream) {
    const float* x0  = (const float*)d_in[0];   // (B, 12)
    const float* u   = (const float*)d_in[1];   // (B, N, 4)
    float* out = (float*)d_out;                 // (B, N, 12)
    const int B = in_sizes[0] / 12;             // 16384
    const int N = in_sizes[1] / (B * 4);        // 256
    dim3 grid(B / BLOCK);
    quad_rollout_kernel<<<grid, BLOCK, 0, stream>>>(x0, u, out, N);
}